// YoloLayer_65609920414053
// MI455X (gfx1250) — compile-verified
//
#include <hip/hip_runtime.h>
#include <hip/hip_bf16.h>
#include <stdint.h>

// ---- problem constants (fixed by the reference) ----
#define TPB   128        // 4 wave32 waves per block
#define CHN   85         // 5 + 80 channels per anchor
#define NCLS  80
#define HWSZ  5776       // 76*76
#define WDIM  76
#define NANCH 3

typedef float v4f __attribute__((ext_vector_type(4)));
typedef __attribute__((address_space(3))) int lds_i;
typedef __attribute__((address_space(1))) int glb_i;

#if defined(__HIP_DEVICE_COMPILE__) && __has_builtin(__builtin_amdgcn_global_load_async_to_lds_b32)
#define USE_ASYNC_LDS 1
#else
#define USE_ASYNC_LDS 0
#endif

__device__ __forceinline__ float fast_exp2(float x) {
#if __has_builtin(__builtin_amdgcn_exp2f)
    return __builtin_amdgcn_exp2f(x);          // v_exp_f32
#else
    return exp2f(x);
#endif
}
__device__ __forceinline__ float fast_rcp(float x) {
#if __has_builtin(__builtin_amdgcn_rcpf)
    return __builtin_amdgcn_rcpf(x);           // v_rcp_f32
#else
    return 1.0f / x;
#endif
}
#define LOG2E 1.44269504088896340736f
__device__ __forceinline__ float fast_exp(float x)     { return fast_exp2(x * LOG2E); }
__device__ __forceinline__ float fast_sigmoid(float x) { return fast_rcp(1.0f + fast_exp(-x)); }

__global__ __launch_bounds__(TPB)
void yolo_head_kernel(const float* __restrict__ in, float* __restrict__ out, int total) {
    // LDS staging tile: [channel][tid] -> bank = tid (conflict-free), 85*128*4 = 43.5 KB
    __shared__ float sm[CHN * TPB];

    const int tid = threadIdx.x;
    int t = blockIdx.x * TPB + tid;
    if (t >= total) t = total - 1;             // shape divides exactly; clamp is idempotent

    const int ba = t / HWSZ;                   // b*3 + a
    const int hw = t - ba * HWSZ;
    const float* gsrc = in + (size_t)ba * (CHN * HWSZ) + hw;

    // ---- stage the 85 channel values of this cell into LDS ----
#if USE_ASYNC_LDS
    {
        lds_i*       dst = (lds_i*)sm + tid;
        const glb_i* src = (const glb_i*)gsrc;
        #pragma unroll
        for (int c = 0; c < CHN; ++c) {
            // per-lane 4B copy; across the wave each channel is a coalesced 128B line
            __builtin_amdgcn_global_load_async_to_lds_b32(
                (glb_i*)(src + (size_t)c * HWSZ), dst + c * TPB, 0, 0);
        }
#if __has_builtin(__builtin_amdgcn_s_wait_asynccnt)
        __builtin_amdgcn_s_wait_asynccnt(0);
#else
        asm volatile("s_wait_asynccnt 0" ::: "memory");
#endif
    }
#else
    #pragma unroll 5
    for (int c = 0; c < CHN; ++c)
        sm[c * TPB + tid] = gsrc[(size_t)c * HWSZ];
#endif
    __syncthreads();   // compiler-level fence; ordering vs async LDS writes

    // ---- box head ----
    const int a  = ba - (ba / NANCH) * NANCH;
    const int gy = hw / WDIM;
    const int gx = hw - gy * WDIM;
    const float invW = 1.0f / 76.0f;

    const float v0 = sm[0 * TPB + tid];
    const float v1 = sm[1 * TPB + tid];
    const float v2 = sm[2 * TPB + tid];
    const float v3 = sm[3 * TPB + tid];
    const float v4 = sm[4 * TPB + tid];

    const float aw = (a == 0) ? 1.5f : ((a == 1) ? 2.375f : 5.0f);
    const float ah = (a == 0) ? 2.0f : ((a == 1) ? 4.5f  : 3.5f);

    v4f box;
    box.x = (fast_sigmoid(v0) + (float)gx) * invW;
    box.y = (fast_sigmoid(v1) + (float)gy) * invW;
    box.z = fast_exp(v2) * aw * invW;
    box.w = fast_exp(v3) * ah * invW;
    const float det = fast_sigmoid(v4);

    __builtin_nontemporal_store(box, (v4f*)out + t);   // streaming store, never re-read

    // ---- stable softmax over 80 classes (exp2 domain) ----
    float m = -3.0e38f;
    #pragma unroll
    for (int c = 0; c < NCLS; ++c)
        m = fmaxf(m, sm[(5 + c) * TPB + tid]);

    float s = 0.0f;
    #pragma unroll
    for (int c = 0; c < NCLS; ++c)
        s += fast_exp2((sm[(5 + c) * TPB + tid] - m) * LOG2E);

    const float scale = det * fast_rcp(s);

    float* cdst = out + (size_t)total * 4 + (size_t)t * NCLS;
    #pragma unroll
    for (int c = 0; c < NCLS; c += 4) {
        v4f e;
        e.x = fast_exp2((sm[(5 + c + 0) * TPB + tid] - m) * LOG2E) * scale;
        e.y = fast_exp2((sm[(5 + c + 1) * TPB + tid] - m) * LOG2E) * scale;
        e.z = fast_exp2((sm[(5 + c + 2) * TPB + tid] - m) * LOG2E) * scale;
        e.w = fast_exp2((sm[(5 + c + 3) * TPB + tid] - m) * LOG2E) * scale;
        __builtin_nontemporal_store(e, (v4f*)(cdst + c));  // 16B aligned: base % 4 dwords == 0
    }
}

extern "C" void kernel_launch(void* const* d_in, const int* in_sizes, int n_in,
                              void* d_out, int out_size, void* d_ws, size_t ws_size,
                              hipStream_t stream) {
    (void)n_in; (void)out_size; (void)d_ws; (void)ws_size;
    const float* in  = (const float*)d_in[0];
    float*       out = (float*)d_out;

    // cells = B * A * H * W = total_input_elems / 85
    const int total  = (int)((long long)in_sizes[0] / CHN);   // 554,496 for the ref shape
    const int blocks = (total + TPB - 1) / TPB;               // 4332, exact

    yolo_head_kernel<<<blocks, TPB, 0, stream>>>(in, out, total);
}